// SNN_Delay_Conv_Small_54107997995106
// MI455X (gfx1250) — compile-verified
//
#include <hip/hip_runtime.h>
#include <hip/hip_bf16.h>
#include <math.h>

typedef __attribute__((ext_vector_type(2))) float v2f;
typedef __attribute__((ext_vector_type(8))) float v8f;

#define NBATCH 64
#define TT 20
#define TP 44      // padded time axis length for every staged input
#define DSTORE 24  // 21 delay taps zero-padded to 24 (divisible by WMMA K-step 4)

// ---------------- workspace offsets (float elements) ----------------
// gauss kernels (M padded to multiple of 16 for conv1/conv3; pre-zeroed)
#define OFF_GK1    0u         // 16 * 1200        = 19200
#define OFF_GK2    19200u     // 32 * 7200        = 230400
#define OFF_GK3    249600u    // 112 * 19200      = 2150400
#define OFF_GK4    2400000u   // 10*100*21        = 21000
#define GK_TOTAL   2421000u
#define OFF_XPAD1  2421000u   // 64*2*32*32*44    = 5767168
#define OFF_Y1     8188168u   // 12 * 1003520     = 12042240   (ends 20230408)
#define OFF_POOL1  2421000u   // 64*12*14*14*20   = 3010560    (reuses xpad1)
#define OFF_STATS1 5431560u   // 2*12*20          = 480
#define OFF_XPAD2  5432040u   // 64*12*196*44     = 6623232    (reuses xpad1 tail + y1 head)
#define OFF_Y2     12055272u  // 32 * 128000      = 4096000
#define OFF_POOL2  2421000u   // 64*32*25*20      = 1024000
#define OFF_STATS2 3445000u   // 2*32*20          = 1280
#define OFF_XPAD3  3446280u   // 64*800*44        = 2252800
#define OFF_Y3     5699080u   // 100*1280         = 128000
#define OFF_STATS3 5827080u   // 2*100*20         = 4000
#define OFF_XPAD4  5831080u   // 64*100*44        = 281600
// peak live = 20,230,408 floats = 80.9 MB of d_ws

__global__ void k_fill0(float* __restrict__ p, long n) {
  long i = (long)blockIdx.x * blockDim.x + threadIdx.x;
  if (i < n) p[i] = 0.f;
}

// DCLS gauss kernel construction: out[s*dstore + d] = W[s] * g_norm(d; P[s])
__global__ void k_gauss(const float* __restrict__ W, const float* __restrict__ P,
                        float* __restrict__ out, int nsyn, int dstore) {
  int s = blockIdx.x * blockDim.x + threadIdx.x;
  if (s >= nsyn) return;
  float c = P[s] + 10.f;              // MAX_DELAY//2 = 10
  const float inv_sig = 1.f / 10.27f; // SIG + 0.27
  float g[21];
  float sum = 0.f;
  #pragma unroll
  for (int d = 0; d < 21; ++d) {
    float z = ((float)d - c) * inv_sig;
    float e = expf(-0.5f * z * z);
    g[d] = e; sum += e;
  }
  float inv = 1.f / (sum + 1e-7f);
  float w = W[s];
  for (int d = 0; d < 21; ++d) out[(size_t)s * dstore + d] = w * g[d] * inv;
  for (int d = 21; d < dstore; ++d) out[(size_t)s * dstore + d] = 0.f;
}

// data (B,T,2,32,32) -> xpad1 (b,ci,h,w,tau) tau in [0,44), left-pad 20 zeros
__global__ void k_xpad1(const float* __restrict__ data, float* __restrict__ xp, long n) {
  long i = (long)blockIdx.x * blockDim.x + threadIdx.x;
  if (i >= n) return;
  int tau = (int)(i % TP); long r = i / TP;
  int w = (int)(r % 32); r /= 32;
  int h = (int)(r % 32); r /= 32;
  int ci = (int)(r % 2);
  int b  = (int)(r / 2);
  float v = 0.f;
  if (tau >= 20 && tau < 40)
    v = data[((((size_t)b * TT + (tau - 20)) * 2 + ci) * 32 + h) * 32 + w];
  xp[i] = v;
}

// Implicit-GEMM delay-conv on the fp32 matrix pipe (V_WMMA_F32_16X16X4_F32).
// Y[m][n] = sum_k A[m][k] * X(k,n);  n=(b,oh,ow,t), k=(ci,kh,kw,d), K-step 4.
// One wave: 16x16 N-tile x MT M-tiles (B fragments reused across M-tiles).
template<int IC,int IH,int IW,int KH,int KW,int OH,int OW,int MT,int MV>
__global__ __launch_bounds__(256) void k_conv_wmma(const float* __restrict__ A,
                                                   const float* __restrict__ X,
                                                   float* __restrict__ Y) {
  constexpr int  SYN = IC * KH * KW;
  constexpr int  K   = SYN * DSTORE;
  constexpr long N   = (long)NBATCH * OH * OW * TT;
  const int lane  = threadIdx.x & 31;
  const int wid   = threadIdx.x >> 5;
  const long ntile = (long)blockIdx.x * 8 + wid;   // grid sized exactly: EXEC all-ones
  const int nlane = lane & 15;
  const int half  = lane >> 4;

  // decode this lane's output column n -> (b, oh, ow, t)
  long n = ntile * 16 + nlane;
  const int t  = (int)(n % TT); long r = n / TT;
  const int ow = (int)(r % OW); r /= OW;
  const int oh = (int)(r % OH);
  const int b  = (int)(r / OH);

  const float* Ab[MT];
  #pragma unroll
  for (int m = 0; m < MT; ++m)
    Ab[m] = A + (size_t)(m * 16 + nlane) * K + 2 * half;

  v8f acc[MT];
  #pragma unroll
  for (int m = 0; m < MT; ++m) acc[m] = (v8f){0.f,0.f,0.f,0.f,0.f,0.f,0.f,0.f};

  for (int syn = 0; syn < SYN; ++syn) {
    const int ci = syn / (KH * KW);
    const int kh = (syn / KW) % KH;
    const int kw = syn % KW;
    const float* Xb = X + ((((size_t)b * IC + ci) * IH + (oh + kh)) * IW + (ow + kw)) * TP
                        + t + 2 * half;
    #pragma unroll
    for (int ds = 0; ds < DSTORE; ds += 4) {
      v2f bf; bf.x = Xb[ds]; bf.y = Xb[ds + 1];          // B[k+2h][n], B[k+2h+1][n]
      #pragma unroll
      for (int m = 0; m < MT; ++m) {
        v2f af = *(const v2f*)(Ab[m] + syn * DSTORE + ds); // A[m][k+2h..+1], 8B aligned
        acc[m] = __builtin_amdgcn_wmma_f32_16x16x4_f32(
            false, af, false, bf, (short)0, acc[m], false, false);
      }
    }
  }

  const long ncol = ntile * 16 + nlane;
  #pragma unroll
  for (int m = 0; m < MT; ++m) {
    #pragma unroll
    for (int rr = 0; rr < 8; ++rr) {
      int row = m * 16 + rr + 8 * half;                  // C/D layout: vgpr rr, half-wave
      if (row < MV) Y[(size_t)row * N + ncol] = acc[m][rr];
    }
  }
}

// y [c][(b,oh,ow,t)] -> pooled (b,c,ph,pw,t), 2x2 spatial max
template<int C,int OH,int OW,int PH,int PW>
__global__ void k_pool(const float* __restrict__ Y, float* __restrict__ Pout) {
  constexpr long TOT = (long)NBATCH * C * PH * PW * TT;
  long i = (long)blockIdx.x * blockDim.x + threadIdx.x;
  if (i >= TOT) return;
  int t = (int)(i % TT); long r = i / TT;
  int w = (int)(r % PW); r /= PW;
  int h = (int)(r % PH); r /= PH;
  int c = (int)(r % C);
  int b = (int)(r / C);
  constexpr long N = (long)NBATCH * OH * OW * TT;
  const float* yc = Y + (size_t)c * N;
  long n00 = (((long)b * OH + 2 * h) * OW + 2 * w) * TT + t;
  float v0 = yc[n00],        v1 = yc[n00 + TT];           // (2w, 2w+1)
  float v2 = yc[n00 + (long)OW * TT], v3 = yc[n00 + (long)OW * TT + TT];
  Pout[i] = fmaxf(fmaxf(v0, v1), fmaxf(v2, v3));
}

// BN stats per (c,t) over (b,hw): biased variance, matching jnp.var
template<int C,int HW>
__global__ void k_stats(const float* __restrict__ X, float* __restrict__ S) {
  __shared__ float s1[256], s2[256];
  int ct = blockIdx.x; int c = ct / TT; int t = ct % TT;
  const int n = NBATCH * HW;
  float a = 0.f, q = 0.f;
  for (int i = threadIdx.x; i < n; i += blockDim.x) {
    int b = i / HW, hw = i % HW;
    float v = X[(((size_t)b * C + c) * HW + hw) * TT + t];
    a += v; q += v * v;
  }
  s1[threadIdx.x] = a; s2[threadIdx.x] = q;
  __syncthreads();
  for (int off = 128; off > 0; off >>= 1) {
    if ((int)threadIdx.x < off) {
      s1[threadIdx.x] += s1[threadIdx.x + off];
      s2[threadIdx.x] += s2[threadIdx.x + off];
    }
    __syncthreads();
  }
  if (threadIdx.x == 0) {
    float mu = s1[0] / n;
    S[ct] = mu;
    S[C * TT + ct] = s2[0] / n - mu * mu;
  }
}

// BN + LIF scan, writing the next layer's time-padded input (incl. zero pad)
template<int C,int HW>
__global__ void k_lif_pool(const float* __restrict__ X, const float* __restrict__ S,
                           const float* __restrict__ gamma, const float* __restrict__ beta,
                           float* __restrict__ XPO) {
  constexpr long TOT = (long)NBATCH * C * HW;
  long i = (long)blockIdx.x * blockDim.x + threadIdx.x;
  if (i >= TOT) return;
  int hw = (int)(i % HW); long r = i / HW;
  int c = (int)(r % C);
  int b = (int)(r / C);
  const float* xin = X + (((size_t)b * C + c) * HW + hw) * TT;
  float* orow = XPO + (((size_t)b * C + c) * HW + hw) * TP;
  #pragma unroll
  for (int tau = 0; tau < 20; ++tau) orow[tau] = 0.f;
  orow[40] = 0.f; orow[41] = 0.f; orow[42] = 0.f; orow[43] = 0.f;
  float g = gamma[c], be = beta[c];
  float mem = 0.f;
  for (int t = 0; t < TT; ++t) {
    float mu = S[c * TT + t], var = S[C * TT + c * TT + t];
    float xn = (xin[t] - mu) * rsqrtf(var + 1e-5f) * g + be;
    float reset = mem > 1.f ? 1.f : 0.f;     // detached subtract reset, THR=1
    mem = 0.9f * mem + xn - reset;
    orow[20 + t] = mem > 1.f ? 1.f : 0.f;    // spike = (mem - THR) > 0
  }
}

__global__ void k_stats_rows(const float* __restrict__ X, float* __restrict__ S) {
  int i = blockIdx.x * blockDim.x + threadIdx.x;
  if (i >= 100 * TT) return;
  int o = i / TT, t = i % TT;
  float a = 0.f, q = 0.f;
  for (int b = 0; b < NBATCH; ++b) {
    float v = X[(size_t)o * (NBATCH * TT) + b * TT + t];
    a += v; q += v * v;
  }
  float mu = a / NBATCH;
  S[i] = mu;
  S[100 * TT + i] = q / NBATCH - mu * mu;
}

__global__ void k_lif_rows(const float* __restrict__ X, const float* __restrict__ S,
                           const float* __restrict__ gamma, const float* __restrict__ beta,
                           float* __restrict__ XPO) {
  int i = blockIdx.x * blockDim.x + threadIdx.x;
  if (i >= NBATCH * 100) return;
  int o = i % 100, b = i / 100;
  const float* xin = X + (size_t)o * (NBATCH * TT) + b * TT;
  float* orow = XPO + ((size_t)b * 100 + o) * TP;
  #pragma unroll
  for (int tau = 0; tau < 20; ++tau) orow[tau] = 0.f;
  orow[40] = 0.f; orow[41] = 0.f; orow[42] = 0.f; orow[43] = 0.f;
  float g = gamma[o], be = beta[o];
  float mem = 0.f;
  for (int t = 0; t < TT; ++t) {
    float mu = S[o * TT + t], var = S[100 * TT + o * TT + t];
    float xn = (xin[t] - mu) * rsqrtf(var + 1e-5f) * g + be;
    float reset = mem > 1.f ? 1.f : 0.f;
    mem = 0.9f * mem + xn - reset;
    orow[20 + t] = mem > 1.f ? 1.f : 0.f;
  }
}

// layer 4: delay-conv1d (10,100,21) + LIF; out = [spk (T,B,10) | mem (T,B,10)]
__global__ void k_layer4(const float* __restrict__ K4, const float* __restrict__ XP,
                         float* __restrict__ out) {
  int i = blockIdx.x * blockDim.x + threadIdx.x;
  if (i >= NBATCH * 10) return;
  int o = i % 10, b = i / 10;
  float mem = 0.f;
  for (int t = 0; t < TT; ++t) {
    float acc = 0.f;
    for (int ci = 0; ci < 100; ++ci) {
      const float* kk = K4 + ((size_t)o * 100 + ci) * 21;
      const float* xr = XP + ((size_t)b * 100 + ci) * TP + t;
      #pragma unroll
      for (int d = 0; d < 21; ++d) acc = fmaf(kk[d], xr[d], acc);
    }
    float reset = mem > 1.f ? 1.f : 0.f;
    mem = 0.9f * mem + acc - reset;
    float spk = mem > 1.f ? 1.f : 0.f;
    out[((size_t)t * NBATCH + b) * 10 + o] = spk;
    out[12800 + ((size_t)t * NBATCH + b) * 10 + o] = mem;
  }
}

extern "C" void kernel_launch(void* const* d_in, const int* in_sizes, int n_in,
                              void* d_out, int out_size, void* d_ws, size_t ws_size,
                              hipStream_t stream) {
  (void)in_sizes; (void)n_in; (void)out_size; (void)ws_size;
  const float* data = (const float*)d_in[0];
  const float* W1 = (const float*)d_in[1];  const float* P1 = (const float*)d_in[2];
  const float* g1 = (const float*)d_in[3];  const float* b1 = (const float*)d_in[4];
  const float* W2 = (const float*)d_in[5];  const float* P2 = (const float*)d_in[6];
  const float* g2 = (const float*)d_in[7];  const float* b2 = (const float*)d_in[8];
  const float* W3 = (const float*)d_in[9];  const float* P3 = (const float*)d_in[10];
  const float* g3 = (const float*)d_in[11]; const float* b3 = (const float*)d_in[12];
  const float* W4 = (const float*)d_in[13]; const float* P4 = (const float*)d_in[14];
  float* ws  = (float*)d_ws;
  float* out = (float*)d_out;

  // zero padded gauss-kernel region (padded M rows / taps must be exactly 0)
  { long n = GK_TOTAL; k_fill0<<<(unsigned)((n + 255) / 256), 256, 0, stream>>>(ws, n); }
  k_gauss<<<(600   + 255) / 256, 256, 0, stream>>>(W1, P1, ws + OFF_GK1, 600,   DSTORE);
  k_gauss<<<(9600  + 255) / 256, 256, 0, stream>>>(W2, P2, ws + OFF_GK2, 9600,  DSTORE);
  k_gauss<<<(80000 + 255) / 256, 256, 0, stream>>>(W3, P3, ws + OFF_GK3, 80000, DSTORE);
  k_gauss<<<(1000  + 255) / 256, 256, 0, stream>>>(W4, P4, ws + OFF_GK4, 1000,  21);

  { long n = 5767168; k_xpad1<<<(unsigned)((n + 255) / 256), 256, 0, stream>>>(data, ws + OFF_XPAD1, n); }

  // conv1: N = 1,003,520 -> 62720 tiles -> 7840 blocks of 8 waves
  k_conv_wmma<2,32,32,5,5,28,28,1,12><<<7840, 256, 0, stream>>>(ws + OFF_GK1, ws + OFF_XPAD1, ws + OFF_Y1);
  { long n = 3010560; k_pool<12,28,28,14,14><<<(unsigned)((n + 255) / 256), 256, 0, stream>>>(ws + OFF_Y1, ws + OFF_POOL1); }
  k_stats<12,196><<<12 * TT, 256, 0, stream>>>(ws + OFF_POOL1, ws + OFF_STATS1);
  { long n = (long)NBATCH * 12 * 196;
    k_lif_pool<12,196><<<(unsigned)((n + 255) / 256), 256, 0, stream>>>(ws + OFF_POOL1, ws + OFF_STATS1, g1, b1, ws + OFF_XPAD2); }

  // conv2: N = 128,000 -> 8000 tiles -> 1000 blocks, 2 M-tiles/wave
  k_conv_wmma<12,14,14,5,5,10,10,2,32><<<1000, 256, 0, stream>>>(ws + OFF_GK2, ws + OFF_XPAD2, ws + OFF_Y2);
  { long n = 1024000; k_pool<32,10,10,5,5><<<(unsigned)((n + 255) / 256), 256, 0, stream>>>(ws + OFF_Y2, ws + OFF_POOL2); }
  k_stats<32,25><<<32 * TT, 256, 0, stream>>>(ws + OFF_POOL2, ws + OFF_STATS2);
  { long n = (long)NBATCH * 32 * 25;
    k_lif_pool<32,25><<<(unsigned)((n + 255) / 256), 256, 0, stream>>>(ws + OFF_POOL2, ws + OFF_STATS2, g2, b2, ws + OFF_XPAD3); }

  // conv3: N = 1280 -> 80 tiles -> 10 blocks, 7 M-tiles/wave (M padded 100->112)
  k_conv_wmma<800,1,1,1,1,1,1,7,100><<<10, 256, 0, stream>>>(ws + OFF_GK3, ws + OFF_XPAD3, ws + OFF_Y3);
  k_stats_rows<<<(2000 + 255) / 256, 256, 0, stream>>>(ws + OFF_Y3, ws + OFF_STATS3);
  k_lif_rows<<<(6400 + 255) / 256, 256, 0, stream>>>(ws + OFF_Y3, ws + OFF_STATS3, g3, b3, ws + OFF_XPAD4);

  k_layer4<<<(640 + 255) / 256, 256, 0, stream>>>(ws + OFF_GK4, ws + OFF_XPAD4, out);
}